// Sampler_14465449853505
// MI455X (gfx1250) — compile-verified
//
#include <hip/hip_runtime.h>
#include <math.h>

// ---------------------------------------------------------------------------
// Masked per-class softmax attention pooling, fused per (n,k) workgroup.
//   N=8, L=16384, C=128, K=16, S=16.  Output (N, K*S, C) fp32.
// Uses V_WMMA_F32_16X16X4_F32 (fp32 matrix core) for both GEMM stages.
// ---------------------------------------------------------------------------

typedef float v2f __attribute__((ext_vector_type(2)));
typedef float v8f __attribute__((ext_vector_type(8)));

#define N_ 8
#define L_ 16384
#define C_ 128
#define K_ 16
#define S_ 16

__device__ __forceinline__ v8f wmma_f32_k4(v2f a, v2f b, v8f c) {
  // 8 args: (neg_a, A, neg_b, B, c_mod, C, reuse_a, reuse_b)
  return __builtin_amdgcn_wmma_f32_16x16x4_f32(
      false, a, false, b, (short)0, c, false, false);
}

__global__ __launch_bounds__(256)
void Sampler_14465449853505_kernel(const float* __restrict__ feat,
                                   const int* __restrict__ cmap,
                                   const float* __restrict__ W,
                                   float* __restrict__ out) {
  const int k    = blockIdx.x;   // class
  const int n    = blockIdx.y;   // batch
  const int tid  = threadIdx.x;
  const int lane = tid & 31;
  const int wv   = tid >> 5;     // wave id 0..7
  const int col  = lane & 15;    // WMMA N / M-within-fragment index
  const int hi   = lane >> 4;    // half-wave select

  __shared__ unsigned short idxbuf[L_];        // 32 KB: member row indices
  __shared__ float ebuf[8][16 * 18];           // per-wave E tile, padded rows
  __shared__ float accbuf[8][S_ * C_];         // 64 KB: per-wave acc partials
  __shared__ float m_part[8][S_];
  __shared__ float d_part[8][S_];
  __shared__ float m_final[S_];
  __shared__ int   s_count;

  const float* feat_n = feat + (size_t)n * L_ * C_;
  const int*   cmap_n = cmap + (size_t)n * L_;

  // ---- Phase 0: deterministic compaction of member indices (wave 0) ------
  if (wv == 0) {
    int count = 0;
    for (int base = 0; base < L_; base += 32) {
      bool hit = (cmap_n[base + lane] == k);
      unsigned mask = (unsigned)__ballot(hit);          // wave32: low 32 bits
      int pre = __popc(mask & ((1u << lane) - 1u));
      if (hit) idxbuf[count + pre] = (unsigned short)(base + lane);
      count += __popc(mask);
    }
    if (lane == 0) s_count = count;
  }
  __syncthreads();
  const int M = s_count;
  const int nChunks = (M + 15) >> 4;
  const int Mc = (M > 0) ? (M - 1) : 0;   // clamp helper

  // ---- Load W[k] as 32 chained fp32 A-fragments (A: 16x4 layout) ---------
  // lane: M=col, a.x = W[k][col][4kk+2hi], a.y = W[k][col][4kk+2hi+1]
  v2f wA[32];
  {
    const float* wrow = W + (size_t)(k * S_ + col) * C_;
#pragma unroll
    for (int kk = 0; kk < 32; ++kk)
      wA[kk] = *(const v2f*)(wrow + 4 * kk + 2 * hi);
  }

  // ---- Pass A: per-slot running max over member logits -------------------
  v8f maxv;
#pragma unroll
  for (int r = 0; r < 8; ++r) maxv[r] = -INFINITY;

  for (int ch = wv; ch < nChunks; ch += 8) {
    const int base = ch << 4;
    const int li   = base + col;
    const int row  = (int)idxbuf[(li < M) ? li : Mc];
    const float* fr = feat_n + (size_t)row * C_;
    v8f d = {0.f, 0.f, 0.f, 0.f, 0.f, 0.f, 0.f, 0.f};
#pragma unroll
    for (int kk = 0; kk < 32; ++kk) {
      v2f b = *(const v2f*)(fr + 4 * kk + 2 * hi);   // B = feat^T fragment
      d = wmma_f32_k4(wA[kk], b, d);
    }
    const bool valid = (li < M);
#pragma unroll
    for (int r = 0; r < 8; ++r)
      maxv[r] = fmaxf(maxv[r], valid ? d[r] : -INFINITY);
  }
  // reduce over the 16 column-lanes of each half-wave
#pragma unroll
  for (int mm = 1; mm < 16; mm <<= 1) {
#pragma unroll
    for (int r = 0; r < 8; ++r)
      maxv[r] = fmaxf(maxv[r], __shfl_xor(maxv[r], mm, 32));
  }
  if (col == 0) {
#pragma unroll
    for (int r = 0; r < 8; ++r) m_part[wv][r + 8 * hi] = maxv[r];
  }
  __syncthreads();
  if (tid < S_) {
    float mg = -INFINITY;
#pragma unroll
    for (int w = 0; w < 8; ++w) mg = fmaxf(mg, m_part[w][tid]);
    m_final[tid] = mg;
  }
  __syncthreads();

  float mvec[8];
#pragma unroll
  for (int r = 0; r < 8; ++r) mvec[r] = m_final[r + 8 * hi];

  // ---- Pass B: exp, denom, weighted accumulation (second WMMA stage) ----
  v8f dsum = {0.f, 0.f, 0.f, 0.f, 0.f, 0.f, 0.f, 0.f};
  v8f acc[8];
#pragma unroll
  for (int t = 0; t < 8; ++t) acc[t] = dsum;

  for (int ch = wv; ch < nChunks; ch += 8) {
    const int base = ch << 4;
    const int li   = base + col;
    const int row  = (int)idxbuf[(li < M) ? li : Mc];
    const float* fr = feat_n + (size_t)row * C_;
    v8f d = {0.f, 0.f, 0.f, 0.f, 0.f, 0.f, 0.f, 0.f};
#pragma unroll
    for (int kk = 0; kk < 32; ++kk) {
      v2f b = *(const v2f*)(fr + 4 * kk + 2 * hi);
      d = wmma_f32_k4(wA[kk], b, d);
    }
    const bool valid = (li < M);
#pragma unroll
    for (int r = 0; r < 8; ++r) {
      float e = valid ? __expf(d[r] - mvec[r]) : 0.0f;
      dsum[r] += e;
      ebuf[wv][(r + 8 * hi) * 18 + col] = e;   // E[s][kd], row pad 18
    }
    // acc[s][c] += sum_kd E[s][kd] * feat[member kd][c]
#pragma unroll
    for (int kk = 0; kk < 4; ++kk) {
      v2f aE = { ebuf[wv][col * 18 + 4 * kk + 2 * hi],
                 ebuf[wv][col * 18 + 4 * kk + 2 * hi + 1] };
      const int p0 = base + 4 * kk + 2 * hi;
      const int p1 = p0 + 1;
      const float* f0 = feat_n + (size_t)idxbuf[(p0 < M) ? p0 : Mc] * C_;
      const float* f1 = feat_n + (size_t)idxbuf[(p1 < M) ? p1 : Mc] * C_;
#pragma unroll
      for (int t = 0; t < 8; ++t) {
        v2f bF = { f0[t * 16 + col], f1[t * 16 + col] };
        acc[t] = wmma_f32_k4(aE, bF, acc[t]);
      }
    }
  }

  // per-slot denominator: sum over the 16 column-lanes
#pragma unroll
  for (int mm = 1; mm < 16; mm <<= 1) {
#pragma unroll
    for (int r = 0; r < 8; ++r) dsum[r] += __shfl_xor(dsum[r], mm, 32);
  }
  if (col == 0) {
#pragma unroll
    for (int r = 0; r < 8; ++r) d_part[wv][r + 8 * hi] = dsum[r];
  }
  // stash per-wave accumulator
#pragma unroll
  for (int t = 0; t < 8; ++t) {
#pragma unroll
    for (int r = 0; r < 8; ++r)
      accbuf[wv][(r + 8 * hi) * C_ + t * 16 + col] = acc[t][r];
  }
  __syncthreads();

  // ---- Final fixed-order combine + divide + store ------------------------
  float* out_nk = out + ((size_t)(n * K_ + k) * S_) * C_;
  for (int i = tid; i < S_ * C_; i += 256) {
    const int s = i >> 7;
    float den = 0.f, a = 0.f;
#pragma unroll
    for (int w = 0; w < 8; ++w) { den += d_part[w][s]; a += accbuf[w][i]; }
    out_nk[i] = (den > 0.f) ? (a / den) : 0.f;
  }
}

extern "C" void kernel_launch(void* const* d_in, const int* in_sizes, int n_in,
                              void* d_out, int out_size, void* d_ws, size_t ws_size,
                              hipStream_t stream) {
  (void)in_sizes; (void)n_in; (void)out_size; (void)d_ws; (void)ws_size;
  const float* feat = (const float*)d_in[0];
  const int*   cmap = (const int*)d_in[1];
  const float* W    = (const float*)d_in[2];
  float* out        = (float*)d_out;

  dim3 grid(K_, N_);   // one workgroup per (class, batch)
  dim3 block(256);     // 8 waves (wave32)
  Sampler_14465449853505_kernel<<<grid, block, 0, stream>>>(feat, cmap, W, out);
}